// JKNet_53291954209003
// MI455X (gfx1250) — compile-verified
//
#include <hip/hip_runtime.h>
#include <hip/hip_bf16.h>

#define NN 50000
#define NE 800000
#define INF 256
#define HID 64
#define CATD 256
#define MTILES (NN / 16)   // 3125, exact

typedef float v2f __attribute__((ext_vector_type(2)));
typedef float v8f __attribute__((ext_vector_type(8)));

// ---------------------------------------------------------------------------
// Degree accumulation (int atomics)
// ---------------------------------------------------------------------------
__global__ void jk_degrees(const int* __restrict__ src, const int* __restrict__ dst,
                           int* __restrict__ deg_s, int* __restrict__ deg_d) {
    int t = blockIdx.x * blockDim.x + threadIdx.x;
    if (t < NE) {
        atomicAdd(&deg_s[src[t]], 1);
        atomicAdd(&deg_d[dst[t]], 1);
    }
}

__global__ void jk_rsqrt(const int* __restrict__ deg_s, const int* __restrict__ deg_d,
                         float* __restrict__ dout_is, float* __restrict__ din_is) {
    int t = blockIdx.x * blockDim.x + threadIdx.x;
    if (t < NN) {
        int a = deg_s[t]; if (a < 1) a = 1;
        int b = deg_d[t]; if (b < 1) b = 1;
        dout_is[t] = rsqrtf((float)a);
        din_is[t]  = rsqrtf((float)b);
    }
}

// ---------------------------------------------------------------------------
// fp32 WMMA GEMM: C[M x 64] = diag(scale) * A[M x K] * W[K x 64] (+ bias)
// One wave computes a 16x64 tile via 4 accumulators of V_WMMA_F32_16X16X4_F32.
// Fragment layout (ISA 7.12.2, 32-bit):
//   A 16x4 : lanes 0-15 hold K={0,1} in v[0],v[1]; lanes 16-31 hold K={2,3}
//   B 4x16 : lanes 0-15 hold rows K0(v0),K1(v1) cols N=lane; lanes 16-31 K2,K3
//   C/D    : VGPR r -> row (r + 8*(lane>=16)), col = lane&15
// ---------------------------------------------------------------------------
__global__ void jk_gemm_wmma(const float* __restrict__ A, int lda,
                             const float* __restrict__ rowscale,
                             const float* __restrict__ W,      // K x 64 row-major
                             const float* __restrict__ bias,   // 64 (may be unused)
                             float* __restrict__ C, int ldc,
                             int K, int mtiles, int use_scale, int use_bias) {
    int wave = blockIdx.x * (blockDim.x >> 5) + (threadIdx.x >> 5);
    if (wave >= mtiles) return;                 // wave-uniform: EXEC stays all-1s
    int lane = threadIdx.x & 31;
    int m  = lane & 15;
    int kh = lane >> 4;                         // 0 or 1
    int row = wave * 16 + m;

    float scale = 1.0f;
    if (use_scale) scale = rowscale[row];

    const float* Arow = A + (size_t)row * lda;

    v8f acc0 = {}, acc1 = {}, acc2 = {}, acc3 = {};

    for (int k = 0; k < K; k += 4) {
        int ka = k + 2 * kh;                    // this lane's pair of K values
        v2f a;
        a.x = Arow[ka]     * scale;
        a.y = Arow[ka + 1] * scale;

        const float* Wk = W + (size_t)ka * 64;  // rows ka, ka+1
        v2f b0, b1, b2, b3;
        b0.x = Wk[m];          b0.y = Wk[64 + m];
        b1.x = Wk[16 + m];     b1.y = Wk[64 + 16 + m];
        b2.x = Wk[32 + m];     b2.y = Wk[64 + 32 + m];
        b3.x = Wk[48 + m];     b3.y = Wk[64 + 48 + m];

        acc0 = __builtin_amdgcn_wmma_f32_16x16x4_f32(false, a, false, b0, (short)0, acc0, false, false);
        acc1 = __builtin_amdgcn_wmma_f32_16x16x4_f32(false, a, false, b1, (short)0, acc1, false, false);
        acc2 = __builtin_amdgcn_wmma_f32_16x16x4_f32(false, a, false, b2, (short)0, acc2, false, false);
        acc3 = __builtin_amdgcn_wmma_f32_16x16x4_f32(false, a, false, b3, (short)0, acc3, false, false);
    }

    float bb0 = 0.f, bb1 = 0.f, bb2 = 0.f, bb3 = 0.f;
    if (use_bias) {
        bb0 = bias[m]; bb1 = bias[16 + m]; bb2 = bias[32 + m]; bb3 = bias[48 + m];
    }
    int crow = wave * 16 + 8 * kh;
    for (int r = 0; r < 8; ++r) {
        float* cp = C + (size_t)(crow + r) * ldc;
        cp[m]      = acc0[r] + bb0;
        cp[16 + m] = acc1[r] + bb1;
        cp[32 + m] = acc2[r] + bb2;
        cp[48 + m] = acc3[r] + bb3;
    }
}

// ---------------------------------------------------------------------------
// Edge scatter-sum: Acc[dst[e], :] += X[src[e], :].  One thread per (edge, f4).
// ---------------------------------------------------------------------------
__global__ void jk_scatter(const float* __restrict__ X, int ldx,
                           const int* __restrict__ src, const int* __restrict__ dst,
                           float* __restrict__ Acc, int ldacc,
                           int shift, long long total) {
    long long t = (long long)blockIdx.x * blockDim.x + threadIdx.x;
    if (t >= total) return;
    int e = (int)(t >> shift);
    int f = ((int)t & ((1 << shift) - 1)) << 2;
    int s = src[e], d = dst[e];
    const float4 v = *(const float4*)(X + (size_t)s * ldx + f);
    float* p = Acc + (size_t)d * ldacc + f;
    __hip_atomic_fetch_add(p + 0, v.x, __ATOMIC_RELAXED, __HIP_MEMORY_SCOPE_AGENT);
    __hip_atomic_fetch_add(p + 1, v.y, __ATOMIC_RELAXED, __HIP_MEMORY_SCOPE_AGENT);
    __hip_atomic_fetch_add(p + 2, v.z, __ATOMIC_RELAXED, __HIP_MEMORY_SCOPE_AGENT);
    __hip_atomic_fetch_add(p + 3, v.w, __ATOMIC_RELAXED, __HIP_MEMORY_SCOPE_AGENT);
}

// ---------------------------------------------------------------------------
// Post-scatter: cat[n, off+f] = relu(H[n,f] * din_is[n] + bias[f])
// ---------------------------------------------------------------------------
__global__ void jk_post(const float* __restrict__ H, const float* __restrict__ din_is,
                        const float* __restrict__ bias, float* __restrict__ Cat,
                        int layer_off) {
    int t = blockIdx.x * blockDim.x + threadIdx.x;   // NN*64 threads
    if (t >= NN * HID) return;
    int n = t >> 6;
    int f = t & 63;
    float v = H[t] * din_is[n] + bias[f];
    Cat[(size_t)n * CATD + layer_off + f] = fmaxf(v, 0.0f);
}

// ---------------------------------------------------------------------------
// Broadcast bias into the output so the final scatter can accumulate in place:
// out[n, f] = b_mlp[f]
// ---------------------------------------------------------------------------
__global__ void jk_bias_init(float* __restrict__ out, const float* __restrict__ bias) {
    int t = blockIdx.x * blockDim.x + threadIdx.x;   // NN*64 threads
    if (t < NN * HID) out[t] = bias[t & 63];
}

// ---------------------------------------------------------------------------
extern "C" void kernel_launch(void* const* d_in, const int* in_sizes, int n_in,
                              void* d_out, int out_size, void* d_ws, size_t ws_size,
                              hipStream_t stream) {
    const float* feat = (const float*)d_in[0];
    const int*   src  = (const int*)d_in[1];
    const int*   dst  = (const int*)d_in[2];
    const float* Wl[4] = {(const float*)d_in[3], (const float*)d_in[5],
                          (const float*)d_in[7], (const float*)d_in[9]};
    const float* bl[4] = {(const float*)d_in[4], (const float*)d_in[6],
                          (const float*)d_in[8], (const float*)d_in[10]};
    const float* Wm = (const float*)d_in[11];
    const float* bm = (const float*)d_in[12];
    float* out = (float*)d_out;

    // ---- workspace carve-out (all 256B aligned) ----
    char* ws = (char*)d_ws;
    auto carve = [&](size_t bytes) -> char* {
        char* p = ws;
        ws += (bytes + 255) & ~(size_t)255;
        return p;
    };
    int*   deg_s   = (int*)  carve((size_t)NN * 4);
    int*   deg_d   = (int*)  carve((size_t)NN * 4);
    float* dout_is = (float*)carve((size_t)NN * 4);
    float* din_is  = (float*)carve((size_t)NN * 4);
    float* xw      = (float*)carve((size_t)NN * HID * 4);   // projected feats
    float* hnext   = (float*)carve((size_t)NN * HID * 4);   // scatter accumulator
    float* cat     = (float*)carve((size_t)NN * CATD * 4);  // JK concat
    float* zmlp    = (float*)carve((size_t)NN * HID * 4);   // cat @ W_mlp

    // ---- degrees ----
    hipMemsetAsync(deg_s, 0, (size_t)NN * 4, stream);
    hipMemsetAsync(deg_d, 0, (size_t)NN * 4, stream);
    jk_degrees<<<(NE + 255) / 256, 256, 0, stream>>>(src, dst, deg_s, deg_d);
    jk_rsqrt<<<(NN + 255) / 256, 256, 0, stream>>>(deg_s, deg_d, dout_is, din_is);

    const int gemm_grid = (MTILES + 3) / 4;   // 4 waves (128 thr) per block

    // ---- 4 GCN layers ----
    for (int i = 0; i < 4; ++i) {
        const float* Ain = (i == 0) ? feat : (cat + (i - 1) * HID);
        int K = (i == 0) ? INF : HID;
        jk_gemm_wmma<<<gemm_grid, 128, 0, stream>>>(
            Ain, (i == 0) ? INF : CATD, dout_is, Wl[i], nullptr,
            xw, HID, K, MTILES, /*use_scale=*/1, /*use_bias=*/0);

        hipMemsetAsync(hnext, 0, (size_t)NN * HID * 4, stream);
        long long tot = (long long)NE * (HID / 4);          // shift = 4
        jk_scatter<<<(int)((tot + 255) / 256), 256, 0, stream>>>(
            xw, HID, src, dst, hnext, HID, 4, tot);

        jk_post<<<(NN * HID + 255) / 256, 256, 0, stream>>>(
            hnext, din_is, bl[i], cat, i * HID);
    }

    // ---- final MLP, projection BEFORE the neighbor sum (linearity):
    //      segsum(cat[src]) @ W_mlp + b == segsum((cat @ W_mlp)[src]) + b ----
    jk_gemm_wmma<<<gemm_grid, 128, 0, stream>>>(
        cat, CATD, dout_is, Wm, nullptr,
        zmlp, HID, CATD, MTILES, /*use_scale=*/0, /*use_bias=*/0);

    // seed d_out with the bias, then accumulate the edge sum directly into it
    jk_bias_init<<<(NN * HID + 255) / 256, 256, 0, stream>>>(out, bm);
    long long tot = (long long)NE * (HID / 4);              // shift = 4
    jk_scatter<<<(int)((tot + 255) / 256), 256, 0, stream>>>(
        zmlp, HID, src, dst, out, HID, 4, tot);
}